// SOM_32830730011468
// MI455X (gfx1250) — compile-verified
//
#include <hip/hip_runtime.h>
#include <hip/hip_bf16.h>

#define SOM_DIM   512
#define SOM_ROWS  65536
#define SOM_EPS   1e-6f
#define SOM_NW    ((size_t)SOM_ROWS * SOM_DIM)
#define GRID_A    512
#define NWAVES_A  4096     // 512 blocks * 8 waves, one 16-row tile per wave

typedef __attribute__((ext_vector_type(2))) float v2f;
typedef __attribute__((ext_vector_type(4))) float v4f;
typedef __attribute__((ext_vector_type(8))) float v8f;

// d_ws layout:
//   [0,     16384)  : float bestVal[4096]
//   [16384, 32768)  : int   bestIdx[4096]
//   [32768, 32784)  : float scal[4] = {bmu_x, bmu_y, rate, 1/(2*sigma^2)}

// ---------------------------------------------------------------------------
// Kernel A: per-row squared distance via V_WMMA_F32_16X16X4_F32 diagonal trick
// One wave per 16-row tile (4096 waves for max memory-level parallelism).
// Lane layout matches the ISA A-matrix 16x4 f32 layout: lanes 0-15 supply
// (M=lane, K=0,1), lanes 16-31 supply (M=lane-16, K=2,3). The same registers
// serve as B = A^T, so D[i][i] accumulates the per-row sum of squares.
// ---------------------------------------------------------------------------
__global__ __launch_bounds__(256) void som_dist_kernel(
    const float* __restrict__ x, const float* __restrict__ w,
    float* __restrict__ wsVal, int* __restrict__ wsIdx)
{
    __shared__ float sx[SOM_DIM];
    const int t = threadIdx.x;
    sx[t]       = x[t];
    sx[t + 256] = x[t + 256];
    __syncthreads();

    const int lane = t & 31;
    const int wv   = blockIdx.x * 8 + (t >> 5);      // wave id 0..4095
    const int koff = (lane < 16) ? 0 : 2;            // K phase for this lane
    const int mrow = lane & 15;                      // row-in-tile for this lane

    const int rowBase = wv * 16;
    const float* wp = w + (size_t)(rowBase + mrow) * SOM_DIM + koff;
    const float* xp = &sx[koff];

    v8f c = {};
    #pragma unroll 16
    for (int k0 = 0; k0 < SOM_DIM; k0 += 4) {
        v2f wv2 = *(const v2f*)(wp + k0);
        v2f xv2 = *(const v2f*)(xp + k0);
        v2f a;
        a.x = xv2.x - wv2.x + SOM_EPS;
        a.y = xv2.y - wv2.y + SOM_EPS;
        // D += A * A^T ; diagonal accumulates per-row sum of squares
        c = __builtin_amdgcn_wmma_f32_16x16x4_f32(
                false, a, false, a, (short)0, c, false, false);
    }

    // Diagonal extraction: row i   -> VGPR i, lane i       (i = 0..7)
    //                      row i+8 -> VGPR i, lane i+24
    float bestVal = 3.4e38f;
    int   bestIdx = 0;
    #pragma unroll
    for (int i = 0; i < 8; ++i) {
        float dA = __shfl(c[i], i, 32);
        float dB = __shfl(c[i], i + 24, 32);
        if (dA < bestVal) { bestVal = dA; bestIdx = rowBase + i; }
        if (dB < bestVal) { bestVal = dB; bestIdx = rowBase + 8 + i; }
    }

    if (lane == 0) { wsVal[wv] = bestVal; wsIdx[wv] = bestIdx; }
}

// ---------------------------------------------------------------------------
// Kernel B: reduce 4096 candidates -> bmu; emit bmu as float + scalars for C
// ---------------------------------------------------------------------------
__global__ __launch_bounds__(256) void som_argmin_kernel(
    const float* __restrict__ wsVal, const int* __restrict__ wsIdx,
    const float* __restrict__ locations, const int* __restrict__ itp,
    float* __restrict__ out_bmu, float* __restrict__ scal)
{
    __shared__ float sv[256];
    __shared__ int   si[256];
    const int t = threadIdx.x;

    float bv = 3.4e38f; int bi = 0x7fffffff;
    for (int j = t; j < NWAVES_A; j += 256) {
        float v = wsVal[j]; int idx = wsIdx[j];
        if (v < bv || (v == bv && idx < bi)) { bv = v; bi = idx; }
    }
    sv[t] = bv; si[t] = bi;
    __syncthreads();

    for (int s = 128; s > 0; s >>= 1) {
        if (t < s) {
            float v = sv[t + s]; int idx = si[t + s];
            if (v < sv[t] || (v == sv[t] && idx < si[t])) { sv[t] = v; si[t] = idx; }
        }
        __syncthreads();
    }

    if (t == 0) {
        const int bmu = si[0];
        out_bmu[0] = (float)bmu;
        const float decay = expf(-(float)itp[0] / 2000.0f);
        const float rate  = 0.5f * decay;
        const float sig   = 128.0f * decay;
        scal[0] = locations[2 * bmu];
        scal[1] = locations[2 * bmu + 1];
        scal[2] = rate;
        scal[3] = 1.0f / (2.0f * sig * sig);
    }
}

// ---------------------------------------------------------------------------
// Kernel C: w' = w + rate * influence(row) * (x - w)
// One wave per row; weights re-read should hit the 192MB L2 (128MiB resident
// from kernel A); output stored non-temporally to avoid evicting them.
// ---------------------------------------------------------------------------
__global__ __launch_bounds__(256) void som_update_kernel(
    const float* __restrict__ x, const float* __restrict__ w,
    const float* __restrict__ locations, const float* __restrict__ scal,
    float* __restrict__ out)
{
    __shared__ float sx[SOM_DIM];
    const int t = threadIdx.x;
    sx[t]       = x[t];
    sx[t + 256] = x[t + 256];
    __syncthreads();

    const int lane = t & 31;
    const int row  = blockIdx.x * 8 + (t >> 5);

    const float bx = scal[0], by = scal[1], rate = scal[2], inv2 = scal[3];
    const float dx = bx - locations[2 * row]     + SOM_EPS;
    const float dy = by - locations[2 * row + 1] + SOM_EPS;
    const float f  = rate * expf(-(dx * dx + dy * dy) * inv2);

    const size_t base = (size_t)row * SOM_DIM;
    #pragma unroll
    for (int cc = 0; cc < 4; ++cc) {
        const int col = cc * 128 + lane * 4;
        v4f wv4 = *(const v4f*)(w + base + col);
        v4f xv4 = *(const v4f*)(&sx[col]);
        v4f o;
        o.x = wv4.x + f * (xv4.x - wv4.x);
        o.y = wv4.y + f * (xv4.y - wv4.y);
        o.z = wv4.z + f * (xv4.z - wv4.z);
        o.w = wv4.w + f * (xv4.w - wv4.w);
        __builtin_nontemporal_store(o, (v4f*)(out + base + col));
    }
}

// ---------------------------------------------------------------------------
extern "C" void kernel_launch(void* const* d_in, const int* in_sizes, int n_in,
                              void* d_out, int out_size, void* d_ws, size_t ws_size,
                              hipStream_t stream) {
    const float* x   = (const float*)d_in[0];
    const float* w   = (const float*)d_in[1];
    const float* loc = (const float*)d_in[2];
    const int*   it  = (const int*)d_in[3];
    float* out = (float*)d_out;

    float* wsVal = (float*)d_ws;
    int*   wsIdx = (int*)((char*)d_ws + 16384);
    float* scal  = (float*)((char*)d_ws + 32768);

    som_dist_kernel<<<GRID_A, 256, 0, stream>>>(x, w, wsVal, wsIdx);
    som_argmin_kernel<<<1, 256, 0, stream>>>(wsVal, wsIdx, loc, it,
                                             out + SOM_NW, scal);
    som_update_kernel<<<8192, 256, 0, stream>>>(x, w, loc, scal, out);
}